// ESAFastBotRGCN_32590211842591
// MI455X (gfx1250) — compile-verified
//
#include <hip/hip_runtime.h>
#include <hip/hip_bf16.h>

#define NN 50000
#define EE 600000
#define DD 128

typedef __attribute__((ext_vector_type(2))) float v2f;
typedef __attribute__((ext_vector_type(8))) float v8f;

__device__ __forceinline__ float lrelu_(float v) { return v > 0.f ? v : 0.01f * v; }

// ---------------------------------------------------------------------------
// Kernel 1: feature projections + concat + leaky relu -> x[N,128]
// cols: [0,25) des | [25,53) tweet | [53,78) num | [78,103) cat | [103,128) new
// 8 nodes per block, des/tweet rows staged in LDS (48KB).
// ---------------------------------------------------------------------------
__global__ __launch_bounds__(256) void feat_proj(
    const float* __restrict__ des, const float* __restrict__ tw,
    const float* __restrict__ np,  const float* __restrict__ cp,
    const float* __restrict__ nf,
    const float* __restrict__ Wd, const float* __restrict__ bd,
    const float* __restrict__ Wt, const float* __restrict__ bt,
    const float* __restrict__ Wn, const float* __restrict__ bn,
    const float* __restrict__ Wc, const float* __restrict__ bc,
    const float* __restrict__ Ww, const float* __restrict__ bw,
    float* __restrict__ x)
{
    __shared__ float sdes[8 * 768];
    __shared__ float stw[8 * 768];
    __shared__ float ssm[8 * 20];
    const int base = blockIdx.x * 8;
    const int tid = threadIdx.x;
    for (int i = tid; i < 8 * 768; i += 256) {
        sdes[i] = des[base * 768 + i];
        stw[i]  = tw[base * 768 + i];
    }
    for (int i = tid; i < 8 * 19; i += 256) {
        int n = i / 19, j = i % 19;
        float v;
        if (j < 7)       v = np[(base + n) * 7 + j];
        else if (j < 18) v = cp[(base + n) * 11 + (j - 7)];
        else             v = nf[base + n];
        ssm[n * 20 + j] = v;
    }
    __syncthreads();
    #pragma unroll
    for (int e = 0; e < 4; ++e) {
        const int idx = tid + 256 * e;   // 8 nodes x 128 cols = 1024
        const int n = idx >> 7, c = idx & 127;
        float s;
        if (c < 25) {
            s = bd[c];
            const float* a = sdes + n * 768;
            for (int k = 0; k < 768; ++k) s = fmaf(a[k], Wd[k * 25 + c], s);
        } else if (c < 53) {
            const int cc = c - 25;
            s = bt[cc];
            const float* a = stw + n * 768;
            for (int k = 0; k < 768; ++k) s = fmaf(a[k], Wt[k * 28 + cc], s);
        } else if (c < 78) {
            const int cc = c - 53;
            s = bn[cc];
            const float* a = ssm + n * 20;
            for (int k = 0; k < 7; ++k) s = fmaf(a[k], Wn[k * 25 + cc], s);
        } else if (c < 103) {
            const int cc = c - 78;
            s = bc[cc];
            const float* a = ssm + n * 20 + 7;
            for (int k = 0; k < 11; ++k) s = fmaf(a[k], Wc[k * 25 + cc], s);
        } else {
            const int cc = c - 103;
            s = fmaf(ssm[n * 20 + 18], Ww[cc], bw[cc]);
        }
        x[(base + n) * 128 + c] = lrelu_(s);
    }
}

// ---------------------------------------------------------------------------
// Kernel 2: C[N,128] = A[N,128] @ B[128,128] with fused epilogue.
//   flags&1 : + bias[col]
//   flags&2 : leaky relu
//   flags&4 : + agg[r,c] / max(deg[r],1)      (RGCN mean-aggregate combine)
// One block = 16 rows; 8 waves, each wave one 16x16 tile via 32 chained
// V_WMMA_F32_16X16X4_F32. A-tile staged in LDS (ds_load_b64 fragments).
// f32 A 16x4 fragment layout: lanes0-15 {K0,K1}, lanes16-31 {K2,K3}.
// ---------------------------------------------------------------------------
__global__ __launch_bounds__(256) void gemm128_wmma(
    const float* __restrict__ A, const float* __restrict__ B,
    const float* __restrict__ bias,
    const float* __restrict__ agg, const float* __restrict__ deg,
    float* __restrict__ out, int flags)
{
    __shared__ float As[16 * 128];
    const int tid = threadIdx.x;
    const int row0 = blockIdx.x * 16;
    {
        const float4* g = (const float4*)(A + row0 * 128);
        float4* s = (float4*)As;
        for (int i = tid; i < 512; i += 256) s[i] = g[i];
    }
    __syncthreads();

    const int lane = tid & 31;
    const int col0 = (tid >> 5) * 16;
    const int half = lane >> 4;          // 0: K lo pair / M 0-7 of D rows, 1: hi
    const int m = lane & 15;

    v8f acc = {0.f, 0.f, 0.f, 0.f, 0.f, 0.f, 0.f, 0.f};
    const float* arow = As + m * 128 + 2 * half;       // {4k+2h, 4k+2h+1}
    const float* bcol = B + 2 * half * 128 + col0 + m; // rows 4k+2h, 4k+2h+1

    #pragma unroll
    for (int k = 0; k < 32; ++k) {
        v2f a = *(const v2f*)(arow + 4 * k);
        v2f b;
        b.x = bcol[(4 * k) * 128];
        b.y = bcol[(4 * k + 1) * 128];
        acc = __builtin_amdgcn_wmma_f32_16x16x4_f32(
            false, a, false, b, (short)0, acc, false, false);
    }

    const int c = col0 + m;
    const float bs = (flags & 1) ? bias[c] : 0.f;
    #pragma unroll
    for (int j = 0; j < 8; ++j) {
        const int r = row0 + j + 8 * half;   // VGPR j: M=j (lanes<16), M=j+8
        float v = acc[j] + bs;
        if (flags & 4) v += agg[r * 128 + c] / fmaxf(deg[r], 1.0f);
        if (flags & 2) v = lrelu_(v);
        out[r * 128 + c] = v;
    }
}

// ---------------------------------------------------------------------------
// Kernel 3: edge gather + atomic scatter-sum + degree count.
// 32 lanes per edge, float4 per lane, hardware f32 atomics.
// ---------------------------------------------------------------------------
__global__ __launch_bounds__(256) void edge_scatter(
    const float* __restrict__ h0, const float* __restrict__ h1,
    const int* __restrict__ ei, const int* __restrict__ et,
    float* __restrict__ agg, float* __restrict__ deg)
{
    const int gid = blockIdx.x * 256 + threadIdx.x;
    const int e = gid >> 5;
    if (e >= EE) return;
    const int lane = gid & 31;
    const int src = ei[e];
    const int dst = ei[EE + e];
    const float* h = (et[e] == 0) ? h0 : h1;
    const float4 v = *(const float4*)(h + src * 128 + lane * 4);
    float* a = agg + dst * 128 + lane * 4;
    __hip_atomic_fetch_add(a + 0, v.x, __ATOMIC_RELAXED, __HIP_MEMORY_SCOPE_AGENT);
    __hip_atomic_fetch_add(a + 1, v.y, __ATOMIC_RELAXED, __HIP_MEMORY_SCOPE_AGENT);
    __hip_atomic_fetch_add(a + 2, v.z, __ATOMIC_RELAXED, __HIP_MEMORY_SCOPE_AGENT);
    __hip_atomic_fetch_add(a + 3, v.w, __ATOMIC_RELAXED, __HIP_MEMORY_SCOPE_AGENT);
    if (lane == 0)
        __hip_atomic_fetch_add(deg + dst, 1.0f, __ATOMIC_RELAXED, __HIP_MEMORY_SCOPE_AGENT);
}

// ---------------------------------------------------------------------------
// Kernel 4: out[N,2] = x @ W2[128,2] + b2. One wave32 per node.
// ---------------------------------------------------------------------------
__global__ __launch_bounds__(256) void out_proj(
    const float* __restrict__ x, const float* __restrict__ W,
    const float* __restrict__ b, float* __restrict__ out)
{
    const int node = blockIdx.x * 8 + (threadIdx.x >> 5);
    const int lane = threadIdx.x & 31;
    const float4 v = *(const float4*)(x + node * 128 + lane * 4);
    const int k = lane * 4;
    float s0 = v.x * W[(k + 0) * 2] + v.y * W[(k + 1) * 2] +
               v.z * W[(k + 2) * 2] + v.w * W[(k + 3) * 2];
    float s1 = v.x * W[(k + 0) * 2 + 1] + v.y * W[(k + 1) * 2 + 1] +
               v.z * W[(k + 2) * 2 + 1] + v.w * W[(k + 3) * 2 + 1];
    #pragma unroll
    for (int off = 16; off > 0; off >>= 1) {
        s0 += __shfl_down(s0, off, 32);
        s1 += __shfl_down(s1, off, 32);
    }
    if (lane == 0) {
        out[node * 2 + 0] = s0 + b[0];
        out[node * 2 + 1] = s1 + b[1];
    }
}

extern "C" void kernel_launch(void* const* d_in, const int* in_sizes, int n_in,
                              void* d_out, int out_size, void* d_ws, size_t ws_size,
                              hipStream_t stream) {
    const float* des  = (const float*)d_in[0];
    const float* tw   = (const float*)d_in[1];
    const float* np   = (const float*)d_in[2];
    const float* cp   = (const float*)d_in[3];
    const float* nf   = (const float*)d_in[4];
    const int*   ei   = (const int*)d_in[5];
    const int*   et   = (const int*)d_in[6];
    const float* Wd = (const float*)d_in[7],  *bd = (const float*)d_in[8];
    const float* Wt = (const float*)d_in[9],  *bt = (const float*)d_in[10];
    const float* Wn = (const float*)d_in[11], *bn = (const float*)d_in[12];
    const float* Wc = (const float*)d_in[13], *bc = (const float*)d_in[14];
    const float* Ww = (const float*)d_in[15], *bw = (const float*)d_in[16];
    const float* Win = (const float*)d_in[17], *bin = (const float*)d_in[18];
    const float* rel1 = (const float*)d_in[19], *root1 = (const float*)d_in[20], *bias1 = (const float*)d_in[21];
    const float* rel2 = (const float*)d_in[22], *root2 = (const float*)d_in[23], *bias2 = (const float*)d_in[24];
    const float* Wo1 = (const float*)d_in[25], *bo1 = (const float*)d_in[26];
    const float* Wo2 = (const float*)d_in[27], *bo2 = (const float*)d_in[28];
    float* out = (float*)d_out;

    // workspace layout (floats): x | h0 | h1 | agg | xb | deg  (~128.2 MB)
    float* x   = (float*)d_ws;
    float* h0  = x   + (size_t)NN * 128;
    float* h1  = h0  + (size_t)NN * 128;
    float* agg = h1  + (size_t)NN * 128;
    float* xb  = agg + (size_t)NN * 128;
    float* deg = xb  + (size_t)NN * 128;

    const dim3 blk(256);
    const dim3 gFeat(NN / 8);       // 6250
    const dim3 gGemm(NN / 16);      // 3125
    const dim3 gEdge(EE / 8);       // 75000 (E*32 threads / 256)
    const dim3 gOut(NN / 8);        // 6250

    // 1) feature projections -> x
    feat_proj<<<gFeat, blk, 0, stream>>>(des, tw, np, cp, nf,
                                         Wd, bd, Wt, bt, Wn, bn, Wc, bc, Ww, bw, x);

    // 2) xb = lrelu(x @ W_in + b_in)
    gemm128_wmma<<<gGemm, blk, 0, stream>>>(x, Win, bin, nullptr, nullptr, xb, 1 | 2);

    // 3) RGCN layer 1: input xb -> output x
    gemm128_wmma<<<gGemm, blk, 0, stream>>>(xb, rel1,          nullptr, nullptr, nullptr, h0, 0);
    gemm128_wmma<<<gGemm, blk, 0, stream>>>(xb, rel1 + 128*128, nullptr, nullptr, nullptr, h1, 0);
    hipMemsetAsync(agg, 0, (size_t)NN * 128 * sizeof(float), stream);
    hipMemsetAsync(deg, 0, (size_t)NN * sizeof(float), stream);
    edge_scatter<<<gEdge, blk, 0, stream>>>(h0, h1, ei, et, agg, deg);
    gemm128_wmma<<<gGemm, blk, 0, stream>>>(xb, root1, bias1, agg, deg, x, 1 | 4);

    // 4) RGCN layer 2: input x -> output xb
    gemm128_wmma<<<gGemm, blk, 0, stream>>>(x, rel2,          nullptr, nullptr, nullptr, h0, 0);
    gemm128_wmma<<<gGemm, blk, 0, stream>>>(x, rel2 + 128*128, nullptr, nullptr, nullptr, h1, 0);
    hipMemsetAsync(agg, 0, (size_t)NN * 128 * sizeof(float), stream);
    hipMemsetAsync(deg, 0, (size_t)NN * sizeof(float), stream);
    edge_scatter<<<gEdge, blk, 0, stream>>>(h0, h1, ei, et, agg, deg);
    gemm128_wmma<<<gGemm, blk, 0, stream>>>(x, root2, bias2, agg, deg, xb, 1 | 4);

    // 5) x = lrelu(xb @ W_out1 + b_out1)
    gemm128_wmma<<<gGemm, blk, 0, stream>>>(xb, Wo1, bo1, nullptr, nullptr, x, 1 | 2);

    // 6) out = x @ W_out2 + b_out2
    out_proj<<<gOut, blk, 0, stream>>>(x, Wo2, bo2, out);
}